// BlockGRU_12481174962905
// MI455X (gfx1250) — compile-verified
//
#include <hip/hip_runtime.h>
#include <hip/hip_bf16.h>
#include <math.h>

// ---------------------------------------------------------------------------
// GRU over 8192 sequential steps, H = I = 512.
//   Phase A (parallel): PRE[g][t][n] = W_g[n, 512:] @ x_t + b_g[n]  via f32 WMMA
//   Phase B (sequential): 16-WGP persistent kernel, recurrent weights resident
//                         in LDS (192 KB/WGP), device-scope barrier per phase.
// Workspace layout (floats): PRE[3*8192*512] | H[512] | RH[512] | CNT (u32)
// Requires ws_size >= 48 MB + 8 KB.
// ---------------------------------------------------------------------------

typedef __attribute__((ext_vector_type(2))) float v2f;
typedef __attribute__((ext_vector_type(8))) float v8f;

#define NT   8192      // total timesteps (B*T)
#define HH   512       // hidden size
#define II   512       // input size
#define WROW 1024      // I + H, row length of each weight matrix
#define NBLK 16        // workgroups in recurrent kernel
#define WPAD 514       // LDS row stride (words): 2*row mod 64 -> conflict-free

// ---------------------------------------------------------------------------
// Init: copy h0 into working H, zero RH and the barrier counter.
// ---------------------------------------------------------------------------
__global__ void gru_init(const float* __restrict__ h0, float* __restrict__ H,
                         float* __restrict__ RH, unsigned* __restrict__ CNT) {
    int i = threadIdx.x;
    if (i < HH) { H[i] = h0[i]; RH[i] = 0.0f; }
    if (i == 0) *CNT = 0u;
}

// ---------------------------------------------------------------------------
// Phase A: PRE[g][t][n] = sum_k X[t][k] * W_g[n][512+k] + b_g[n]
// One wave computes a 16(t) x 64(n) tile: 4 accumulators share the A operand.
// f32 WMMA 16x16x4:
//   A (16x4): lane L holds rows m=L&15, K pair {2*(L>>4), 2*(L>>4)+1} in v0/v1
//   B (4x16): lane L holds col n=L&15, same K-pair striping
//   C/D     : vgpr i holds m = i + 8*(L>>4), n = L&15
// ---------------------------------------------------------------------------
__global__ void gru_pre_gemm(const float* __restrict__ X,
                             const float* __restrict__ Wr, const float* __restrict__ br,
                             const float* __restrict__ Wz, const float* __restrict__ bz,
                             const float* __restrict__ Wh, const float* __restrict__ bh,
                             float* __restrict__ PRE) {
    const int wave = blockIdx.x * (blockDim.x >> 5) + (threadIdx.x >> 5);
    const int lane = threadIdx.x & 31;

    const int g  = wave / (512 * 8);        // gate 0..2
    const int rm = wave % (512 * 8);
    const int tt = rm >> 3;                 // t-tile 0..511 (16 steps each)
    const int ng = rm & 7;                  // n-group 0..7  (64 outputs each)

    const float* W;  const float* bias;
    if (g == 0)      { W = Wr; bias = br; }
    else if (g == 1) { W = Wz; bias = bz; }
    else             { W = Wh; bias = bh; }

    const int m    = lane & 15;
    const int half = lane >> 4;
    const int trow = tt * 16 + m;
    const float* arow  = X + (size_t)trow * II + 2 * half;
    const int    nbase = ng * 64 + (lane & 15);

    v8f acc0 = {}, acc1 = {}, acc2 = {}, acc3 = {};

    for (int kk = 0; kk < II; kk += 4) {
        v2f a;
        a.x = arow[kk];
        a.y = arow[kk + 1];
        #pragma unroll
        for (int s = 0; s < 4; ++s) {
            const int n = nbase + s * 16;
            const float* bp = W + (size_t)n * WROW + HH + kk + 2 * half;
            v2f b;
            b.x = bp[0];
            b.y = bp[1];
            v8f* accp = (s == 0) ? &acc0 : (s == 1) ? &acc1 : (s == 2) ? &acc2 : &acc3;
            *accp = __builtin_amdgcn_wmma_f32_16x16x4_f32(
                /*neg_a=*/false, a, /*neg_b=*/false, b,
                /*c_mod=*/(short)0, *accp, /*reuse_a=*/false, /*reuse_b=*/false);
        }
    }

    float* outg = PRE + (size_t)g * NT * HH;
    #pragma unroll
    for (int s = 0; s < 4; ++s) {
        const int n  = nbase + s * 16;
        const float bv = bias[n];
        const v8f acc = (s == 0) ? acc0 : (s == 1) ? acc1 : (s == 2) ? acc2 : acc3;
        #pragma unroll
        for (int i = 0; i < 8; ++i) {
            const int to = tt * 16 + i + 8 * half;     // output timestep
            outg[(size_t)to * HH + n] = acc[i] + bv;
        }
    }
}

// ---------------------------------------------------------------------------
// Phase B: 16 persistent workgroups (one per WGP), 256 threads (8 waves).
// Block w owns hidden rows [w*32, w*32+32) of every gate; the three 32x512
// recurrent weight slices live in LDS (padded stride 514 -> bank-conflict-free
// since (514*row) % 64 = 2*row is unique over row=0..31).
// Cross-WGP sync: monotonically increasing device-scope atomic counter.
// ---------------------------------------------------------------------------
__device__ __forceinline__ void grid_barrier(unsigned* CNT, unsigned target, int tid) {
    __syncthreads();
    if (tid == 0) {
        __hip_atomic_fetch_add(CNT, 1u, __ATOMIC_ACQ_REL, __HIP_MEMORY_SCOPE_AGENT);
        while (__hip_atomic_load(CNT, __ATOMIC_ACQUIRE, __HIP_MEMORY_SCOPE_AGENT) < target)
            __builtin_amdgcn_s_sleep(2);
    }
    __syncthreads();
}

__global__ void gru_recurrence(const float* __restrict__ Wr,
                               const float* __restrict__ Wz,
                               const float* __restrict__ Wh,
                               const float* __restrict__ PRE,
                               float* H, float* RH, unsigned* CNT,
                               float* __restrict__ out) {
    extern __shared__ float lds[];
    float* wrL  = lds;                    // 32*514
    float* wzL  = wrL + 32 * WPAD;        // 32*514
    float* whL  = wzL + 32 * WPAD;        // 32*514
    float* redA = whL + 32 * WPAD;        // 8*32
    float* redB = redA + 256;             // 8*32
    float* zbuf = redB + 256;             // 32
    float* hbuf = zbuf + 32;              // 32

    const int blk = blockIdx.x;           // 0..15
    const int tid = threadIdx.x;          // 0..255
    const int row = tid & 31;             // lane within wave = hidden row
    const int sub = tid >> 5;             // wave id = K chunk (64 each)
    const int kb  = sub * 64;

    // One-time load of this block's recurrent weight slices (h-columns only).
    for (int idx = tid; idx < 32 * HH; idx += 256) {
        const int r = idx >> 9, k = idx & 511;
        const size_t grow = (size_t)(blk * 32 + r) * WROW;
        wrL[r * WPAD + k] = Wr[grow + k];
        wzL[r * WPAD + k] = Wz[grow + k];
        whL[r * WPAD + k] = Wh[grow + k];
    }
    __syncthreads();

    const float* PRE_r = PRE;
    const float* PRE_z = PRE + (size_t)NT * HH;
    const float* PRE_h = PRE + (size_t)2 * NT * HH;

    unsigned target = 0;
    for (int t = 0; t < NT; ++t) {
        // Prefetch next step's pre-activation slice for this block (hides HBM latency).
        if (tid < 3 && t + 1 < NT) {
            const float* pg = PRE + (size_t)tid * NT * HH + (size_t)(t + 1) * HH + blk * 32;
            __builtin_prefetch(pg, 0, 1);
        }

        // ---- Phase 1: r, z slices --------------------------------------
        float pr = 0.0f, pz = 0.0f;
        for (int j = 0; j < 64; ++j) {
            const int k = kb + j;
            const float hk = H[k];
            pr = __builtin_fmaf(wrL[row * WPAD + k], hk, pr);
            pz = __builtin_fmaf(wzL[row * WPAD + k], hk, pz);
        }
        redA[sub * 32 + row] = pr;
        redB[sub * 32 + row] = pz;
        __syncthreads();
        if (tid < 32) {
            float sr = 0.0f, sz = 0.0f;
            #pragma unroll
            for (int s = 0; s < 8; ++s) { sr += redA[s * 32 + tid]; sz += redB[s * 32 + tid]; }
            const int gr = blk * 32 + tid;
            sr += PRE_r[(size_t)t * HH + gr];
            sz += PRE_z[(size_t)t * HH + gr];
            const float rg  = 1.0f / (1.0f + __expf(-sr));
            const float zg  = 1.0f / (1.0f + __expf(-sz));
            const float h0v = H[gr];
            zbuf[tid] = zg;
            hbuf[tid] = h0v;
            RH[gr] = rg * h0v;
            __threadfence();
        }
        target += NBLK;
        grid_barrier(CNT, target, tid);

        // ---- Phase 2: candidate + update -------------------------------
        float ph = 0.0f;
        for (int j = 0; j < 64; ++j) {
            const int k = kb + j;
            ph = __builtin_fmaf(whL[row * WPAD + k], RH[k], ph);
        }
        redA[sub * 32 + row] = ph;
        __syncthreads();
        if (tid < 32) {
            float sh = 0.0f;
            #pragma unroll
            for (int s = 0; s < 8; ++s) sh += redA[s * 32 + tid];
            const int gr = blk * 32 + tid;
            sh += PRE_h[(size_t)t * HH + gr];
            const float cand = tanhf(sh);
            const float zg = zbuf[tid], h0v = hbuf[tid];
            H[gr] = (1.0f - zg) * h0v + zg * cand;
            __threadfence();
        }
        target += NBLK;
        grid_barrier(CNT, target, tid);
    }

    // Reference returns (h_final, h_final): write both copies.
    if (tid < 32) {
        const int gr = blk * 32 + tid;
        const float v = H[gr];
        out[gr] = v;
        out[HH + gr] = v;
    }
}

// ---------------------------------------------------------------------------
extern "C" void kernel_launch(void* const* d_in, const int* in_sizes, int n_in,
                              void* d_out, int out_size, void* d_ws, size_t ws_size,
                              hipStream_t stream) {
    const float* X  = (const float*)d_in[0];   // embeddings [4,2048,512]
    const float* h0 = (const float*)d_in[1];   // hidden [512]
    const float* Wr = (const float*)d_in[2];
    const float* br = (const float*)d_in[3];
    const float* Wz = (const float*)d_in[4];
    const float* bz = (const float*)d_in[5];
    const float* Wh = (const float*)d_in[6];
    const float* bh = (const float*)d_in[7];
    float* out = (float*)d_out;

    float*    PRE = (float*)d_ws;                       // 3*8192*512 floats (48 MB)
    float*    H   = PRE + (size_t)3 * NT * HH;
    float*    RH  = H + HH;
    unsigned* CNT = (unsigned*)(RH + HH);

    gru_init<<<1, 512, 0, stream>>>(h0, H, RH, CNT);

    // 3 gates * 512 t-tiles * 8 n-groups = 12288 waves; 8 waves / block.
    gru_pre_gemm<<<1536, 256, 0, stream>>>(X, Wr, br, Wz, bz, Wh, bh, PRE);

    // 192 KB weight slices + reduction scratch per workgroup (gfx1250: 320 KB LDS/WGP).
    const size_t lds_bytes = (size_t)(3 * 32 * WPAD + 2 * 256 + 64) * sizeof(float);
    gru_recurrence<<<NBLK, 256, lds_bytes, stream>>>(Wr, Wz, Wh, PRE, H, RH, CNT, out);
}